// HealpixHaarPyramid_68264210203010
// MI455X (gfx1250) — compile-verified
//
#include <hip/hip_runtime.h>
#include <hip/hip_bf16.h>

typedef float v2f __attribute__((ext_vector_type(2)));
typedef float v8f __attribute__((ext_vector_type(8)));

// Haar 4x4 sign: W[r][c] = 0.5 * sgn(r,c); rows of W are Hadamard rows with the
// 2-bit row index bit-reversed (W rows: ++++, ++--, +-+-, +--+).
__device__ __forceinline__ float wsgn(int r, int c) {
    int rr = ((r & 1) << 1) | ((r >> 1) & 1);
    return (__popc(rr & c) & 1) ? -1.0f : 1.0f;
}

// Combined two-level Haar transform of 16 contiguous elements: out = M * in.
// Row 0: grand average (0.25 * all-ones)
// Rows 1..3: level-2 details (0.25 * sign over quad index)
// Rows 4..15: level-1 details, row 4 + 3q + (j-1): 0.5*W[j][k] on quad q only.
__device__ __forceinline__ float haarM(int m, int k) {
    int q = k >> 2, kk = k & 3;
    if (m < 4) return 0.25f * wsgn(m, q);
    int qm = (m - 4) / 3;
    int j  = (m - 4) % 3 + 1;
    return (q == qm) ? 0.5f * wsgn(j, kk) : 0.0f;
}

// One fused two-level Haar pass. Each wave handles one tile = 16 groups of 16
// contiguous fp32 (256 elems) of one batch row:
//   D(16x16) = M(16x16) @ X(16x16), X column n = group n.
// Outputs per tile (all contiguous in their destinations):
//   fine details  (rows 4..15) -> 192 floats at  z + off_fine  + tile*192
//   coarse details(rows 1..3)  ->  48 floats at  z + off_coarse+ tile*48
//   averages      (row 0)      ->  16 floats at  vnext         + tile*16
__global__ __launch_bounds__(256) void haar2lvl_wmma(
    const float* __restrict__ vin, long vin_stride,
    float* __restrict__ z, long z_stride, long off_fine, long off_coarse,
    float* __restrict__ vnext, long vnext_stride,
    int tiles_per_batch, int total_tiles)
{
    __shared__ __align__(16) float sIn[8][256];
    __shared__ __align__(16) float sOut[8][256];

    const int lane = threadIdx.x & 31;
    const int w    = threadIdx.x >> 5;
    const int half = lane >> 4;   // lane half selects K pairing per ISA layout
    const int lm   = lane & 15;

    // A operand = M, split into 4 K-slices of 16x4 each.
    // V_WMMA_F32_16X16X4_F32 A layout: lanes 0-15 hold (M=lane, K={0,1});
    // lanes 16-31 hold (M=lane-16, K={2,3}) across the 2 VGPRs.
    v2f a[4];
#pragma unroll
    for (int s = 0; s < 4; ++s) {
        a[s].x = haarM(lm, 4 * s + 2 * half + 0);
        a[s].y = haarM(lm, 4 * s + 2 * half + 1);
    }

    float* myIn  = sIn[w];
    float* myOut = sOut[w];

    const int waveId = blockIdx.x * 8 + w;
    const int nWaves = gridDim.x * 8;

    for (int t = waveId; t < total_tiles; t += nWaves) {   // wave-uniform
        const int b  = t / tiles_per_batch;
        const int gb = t - b * tiles_per_batch;

        const float* src = vin + (long)b * vin_stride + (long)gb * 256;
        // Coalesced 1 KB load into per-wave LDS stage (no barrier needed:
        // producer and consumer are the same wave; DScnt ordering applies).
        *(float4*)(myIn + lane * 4)       = *(const float4*)(src + lane * 4);
        *(float4*)(myIn + 128 + lane * 4) = *(const float4*)(src + 128 + lane * 4);

        // B operand slice s: B[k][n] = group_n[4s+k].
        // B layout mirrors A: VGPR0 = rows K=0 (lanes 0-15) / K=2 (lanes 16-31).
        v8f c = {};
#pragma unroll
        for (int s = 0; s < 4; ++s) {
            v2f bv = *(const v2f*)(myIn + lm * 16 + 4 * s + 2 * half);
            c = __builtin_amdgcn_wmma_f32_16x16x4_f32(
                    false, a[s], false, bv, (short)0, c, false, false);
        }

        // Scatter D into LDS so global stores are contiguous:
        // [0..191]   fine details, group-major: n*12 + (m-4)
        // [192..239] coarse details: n*3 + (m-1)
        // [240..255] averages: n
        // D layout: VGPR r, lane L -> D[m = r + 8*(L>>4)][n = L&15]
#pragma unroll
        for (int r = 0; r < 8; ++r) {
            int m = r + 8 * half;
            int addr;
            if (m == 0)       addr = 240 + lm;
            else if (m < 4)   addr = 192 + lm * 3 + (m - 1);
            else              addr = lm * 12 + (m - 4);
            myOut[addr] = c[r];
        }

        float* d1g = z + (long)b * z_stride + off_fine   + (long)gb * 192;
        float* d2g = z + (long)b * z_stride + off_coarse + (long)gb * 48;
        float* vng = vnext + (long)b * vnext_stride + (long)gb * 16;

        *(float4*)(d1g + lane * 4)       = *(const float4*)(myOut + lane * 4);        // 128
        *(float2*)(d1g + 128 + lane * 2) = *(const float2*)(myOut + 128 + lane * 2);  //  64
        if (lane < 24)
            *(float2*)(d2g + lane * 2)   = *(const float2*)(myOut + 192 + lane * 2);  //  48
        if (lane < 16)
            vng[lane] = myOut[240 + lane];                                            //  16
    }
}

// Final single level: 12 parents per batch from the 48 remaining averages,
// plus the logdet = 0 tail of the output tuple.
__global__ void haar_final(const float* __restrict__ v8,
                           float* __restrict__ z,
                           float* __restrict__ logdet)
{
    int i = blockIdx.x * blockDim.x + threadIdx.x;
    if (i < 16 * 12) {
        int b = i / 12, p = i % 12;
        const float* s = v8 + b * 48 + p * 4;
        float x0 = s[0], x1 = s[1], x2 = s[2], x3 = s[3];
        float* zr = z + (long)b * 3145728L;
        zr[p]              = 0.5f * (x0 + x1 + x2 + x3);   // 12 coarsest averages
        zr[12 + p * 3 + 0] = 0.5f * (x0 + x1 - x2 - x3);   // level-9 details
        zr[12 + p * 3 + 1] = 0.5f * (x0 - x1 + x2 - x3);
        zr[12 + p * 3 + 2] = 0.5f * (x0 - x1 - x2 + x3);
    }
    if (i < 16) logdet[i] = 0.0f;
}

extern "C" void kernel_launch(void* const* d_in, const int* in_sizes, int n_in,
                              void* d_out, int out_size, void* d_ws, size_t ws_size,
                              hipStream_t stream) {
    (void)in_sizes; (void)n_in; (void)out_size; (void)ws_size;
    const float* x = (const float*)d_in[0];
    float* z      = (float*)d_out;                 // (16, 3145728) flattened
    float* logdet = z + (size_t)16 * 3145728;      // tuple tail: 16 zeros

    // Workspace: average pyramids between fused passes.
    float* v2 = (float*)d_ws;                        // 16 x 196608
    float* v4 = v2 + (size_t)16 * 196608;            // 16 x 12288
    float* v6 = v4 + (size_t)16 * 12288;             // 16 x 768
    float* v8 = v6 + (size_t)16 * 768;               // 16 x 48

    const long N = 3145728;
    // Detail block offsets inside a z row (coarse -> fine after the 12 avgs):
    // off9=12, off8=48, off7=192, off6=768, off5=3072, off4=12288,
    // off3=49152, off2=196608, off1=786432.

    auto launch = [&](const float* vin, long vstride,
                      long offF, long offC,
                      float* vn, long vnstride, int tpb) {
        int total  = 16 * tpb;
        int blocks = (total + 7) / 8;
        if (blocks > 8192) blocks = 8192;
        haar2lvl_wmma<<<blocks, 256, 0, stream>>>(
            vin, vstride, z, N, offF, offC, vn, vnstride, tpb, total);
    };

    launch(x,  N,      786432, 196608, v2, 196608, 12288); // levels 1,2
    launch(v2, 196608,  49152,  12288, v4,  12288,   768); // levels 3,4
    launch(v4, 12288,    3072,    768, v6,    768,    48); // levels 5,6
    launch(v6, 768,       192,     48, v8,     48,     3); // levels 7,8
    haar_final<<<1, 256, 0, stream>>>(v8, z, logdet);      // level 9 + logdet
}